// EsmContactPredictionHead_46153718563094
// MI455X (gfx1250) — compile-verified
//
#include <hip/hip_runtime.h>
#include <math.h>

// Problem constants (B=1, LAYERS=33, HEADS=20, SEQ=512)
#define S   512     // full sequence
#define SC  510     // cropped size (1..510 of memory index space)
#define F   660     // LAYERS*HEADS
#define NFG 20      // feature groups in pass1
#define FPG 33      // features per group (20*33 = 660)

typedef __attribute__((ext_vector_type(2))) float v2f;
typedef __attribute__((ext_vector_type(4))) float v4f;
typedef __attribute__((ext_vector_type(8))) float v8f;

// ---------------------------------------------------------------- zero init
__global__ __launch_bounds__(256) void zero_ws_kernel(float* __restrict__ p, int n) {
  int i = blockIdx.x * 256 + threadIdx.x;
  int stride = gridDim.x * 256;
  for (; i < n; i += stride) p[i] = 0.f;
}

// ---------------------------------------------------------------- pass 1
// One pass over the 692MB tensor. Tile = 32 memory-rows x 128 memory-cols.
// blockIdx.x in [0,64): 16 row-tiles x 4 col-tiles. blockIdx.y = feature group.
// Attention loads are NON-TEMPORAL: the tensor is streamed exactly once, so
// keep the hot 2.4MB atomic accumulators (Wsym, Rg) resident in L2 instead.
// Produces:
//   Wsym[i,j] += w_f * (att[i,j] + att[j,i])   (atomic, symmetric scatter)
//   Rg[f][i]  += rowsum_f(i) + colsum_f(i)     (atomic)
__global__ __launch_bounds__(256) void pass1_kernel(
    const int* __restrict__ tokens, const float* __restrict__ att,
    const float* __restrict__ weight, float* __restrict__ Wsym,
    float* __restrict__ Rg)
{
  const int t    = threadIdx.x;
  const int lane = t & 31;
  const int wv   = t >> 5;           // wave id 0..7
  const int tile = blockIdx.x;
  const int ti   = tile >> 2;        // 0..15 row tile
  const int tj   = tile & 3;         // 0..3  col tile (128 wide)
  const int mc0  = tj * 128 + lane * 4;   // memory column base (16B aligned)

  // column masks (memory col mc maps to cropped col mc-1)
  float mj[4];
#pragma unroll
  for (int c = 0; c < 4; ++c) {
    int mc = mc0 + c;
    mj[c] = (mc >= 1 && mc <= SC && tokens[mc] != 2) ? 1.f : 0.f;
  }
  // row masks: this thread owns memory rows ti*32 + wv + 8*s
  int   mr_[4];
  float mi[4];
#pragma unroll
  for (int s = 0; s < 4; ++s) {
    int mr = ti * 32 + wv + 8 * s;
    mr_[s] = mr;
    mi[s] = (mr >= 1 && mr <= SC && tokens[mr] != 2) ? 1.f : 0.f;
  }

  __shared__ float lcol[8][128];

  float wacc[4][4];
#pragma unroll
  for (int s = 0; s < 4; ++s)
#pragma unroll
    for (int c = 0; c < 4; ++c) wacc[s][c] = 0.f;

  const int f0 = blockIdx.y * FPG;
  for (int fi = 0; fi < FPG; ++fi) {
    const int f = f0 + fi;
    const float wf = weight[f];
    const float* __restrict__ base = att + (size_t)f * S * S;

    float colp0 = 0.f, colp1 = 0.f, colp2 = 0.f, colp3 = 0.f;
#pragma unroll
    for (int s = 0; s < 4; ++s) {
      // streamed-once data: non-temporal (TH=NT) b128 load
      const v4f raw = __builtin_nontemporal_load(
          (const v4f*)(base + (size_t)mr_[s] * S + mc0));
      const float mrow = mi[s];
      float v0 = raw.x * (mrow * mj[0]);
      float v1 = raw.y * (mrow * mj[1]);
      float v2 = raw.z * (mrow * mj[2]);
      float v3 = raw.w * (mrow * mj[3]);

      wacc[s][0] += wf * v0; wacc[s][1] += wf * v1;
      wacc[s][2] += wf * v2; wacc[s][3] += wf * v3;
      colp0 += v0; colp1 += v1; colp2 += v2; colp3 += v3;

      // full row (128 cols) lives in this wave -> butterfly reduce
      float rowv = (v0 + v1) + (v2 + v3);
#pragma unroll
      for (int off = 16; off > 0; off >>= 1) rowv += __shfl_xor(rowv, off, 32);
      if (lane == 0 && mrow != 0.f)
        atomicAdd(&Rg[(size_t)f * S + (mr_[s] - 1)], rowv);
    }

    // cross-wave column reduce via LDS (each wave covered 4 of the 32 rows)
    lcol[wv][lane * 4 + 0] = colp0;
    lcol[wv][lane * 4 + 1] = colp1;
    lcol[wv][lane * 4 + 2] = colp2;
    lcol[wv][lane * 4 + 3] = colp3;
    __syncthreads();
    if (t < 128) {
      float cs = lcol[0][t];
#pragma unroll
      for (int w = 1; w < 8; ++w) cs += lcol[w][t];
      int mc = tj * 128 + t;
      if (mc >= 1 && mc <= SC)
        atomicAdd(&Rg[(size_t)f * S + (mc - 1)], cs);
    }
    __syncthreads();
  }

  // flush weighted accumulation symmetrically: Wsym gets both (i,j) and (j,i)
#pragma unroll
  for (int s = 0; s < 4; ++s) {
    int gi = mr_[s] - 1;
    if (gi < 0 || gi >= SC) continue;
#pragma unroll
    for (int c = 0; c < 4; ++c) {
      int gj = mc0 + c - 1;
      if (gj < 0 || gj >= SC) continue;
      float v = wacc[s][c];
      atomicAdd(&Wsym[(size_t)gi * S + gj], v);
      atomicAdd(&Wsym[(size_t)gj * S + gi], v);
    }
  }
}

// ---------------------------------------------------------------- mid
// Per feature f: T_f = sum_i r_f[i] (== a12), U[f][i] = r_f[i] * w_f / T_f
__global__ __launch_bounds__(256) void mid_kernel(
    const float* __restrict__ Rg, const float* __restrict__ weight,
    float* __restrict__ Ug)
{
  const int f = blockIdx.x;
  const int t = threadIdx.x;
  const float r0 = Rg[(size_t)f * S + t];
  const float r1 = Rg[(size_t)f * S + t + 256];
  float s = r0 + r1;
#pragma unroll
  for (int off = 16; off > 0; off >>= 1) s += __shfl_xor(s, off, 32);
  __shared__ float ws[8];
  if ((t & 31) == 0) ws[t >> 5] = s;
  __syncthreads();
  float T = ws[0];
#pragma unroll
  for (int w = 1; w < 8; ++w) T += ws[w];
  const float scale = weight[f] / T;   // reference has no zero-guard either
  Ug[(size_t)f * S + t]       = r0 * scale;
  Ug[(size_t)f * S + t + 256] = r1 * scale;
}

// ---------------------------------------------------------------- pass 2
// C = Uᵀ(510x660) x R(660x510) via V_WMMA_F32_16X16X4_F32, fused with
// out = sigmoid(Wsym - C + bias). One wave per 16x16 output tile.
__global__ __launch_bounds__(256) void pass2_kernel(
    const float* __restrict__ Ug, const float* __restrict__ Rg,
    const float* __restrict__ Wsym, const float* __restrict__ bias,
    float* __restrict__ out)
{
  const int t    = threadIdx.x;
  const int lane = t & 31;
  const int wv   = t >> 5;
  const int wave = blockIdx.x * 8 + wv;   // 0..1023
  const int ti = wave >> 5, tj = wave & 31;
  const int i0 = ti * 16, j0 = tj * 16;
  const int half = lane >> 4;             // K sub-pair selector per ISA layout
  const int lp   = lane & 15;

  v8f acc = {0.f, 0.f, 0.f, 0.f, 0.f, 0.f, 0.f, 0.f};

  // A[m][k] = Ug[k][m] (m = i0+lp), B[k][n] = Rg[k][n] (n = j0+lp)
  const float* __restrict__ Ap = Ug + (size_t)(2 * half) * S + i0 + lp;
  const float* __restrict__ Bp = Rg + (size_t)(2 * half) * S + j0 + lp;
  for (int k0 = 0; k0 < F; k0 += 4) {
    v2f a, b;
    a.x = Ap[0]; a.y = Ap[S];      // K = k0+2*half, k0+2*half+1
    b.x = Bp[0]; b.y = Bp[S];
    acc = __builtin_amdgcn_wmma_f32_16x16x4_f32(
        false, a, false, b, (short)0, acc, false, false);
    Ap += 4 * S; Bp += 4 * S;
  }

  const float bb = bias[0];
#pragma unroll
  for (int r = 0; r < 8; ++r) {
    int i = i0 + r + 8 * half;   // C/D layout: VGPR r, lanes16-31 -> M+8
    int j = j0 + lp;
    if (i < SC && j < SC) {
      float val = Wsym[(size_t)i * S + j] - acc[r] + bb;
      float sg = 1.f / (1.f + __expf(-val));
      // output is written once and never re-read: non-temporal store
      __builtin_nontemporal_store(sg, &out[(size_t)i * SC + j]);
    }
  }
}

// ---------------------------------------------------------------- launcher
extern "C" void kernel_launch(void* const* d_in, const int* in_sizes, int n_in,
                              void* d_out, int out_size, void* d_ws, size_t ws_size,
                              hipStream_t stream) {
  const int*   tokens = (const int*)d_in[0];    // (1,512) int32
  const float* att    = (const float*)d_in[1];  // (1,33,20,512,512) f32
  const float* weight = (const float*)d_in[2];  // (660,) f32
  const float* bias   = (const float*)d_in[3];  // (1,) f32
  float* out = (float*)d_out;                   // (1,510,510) f32

  // workspace layout (floats): Wsym 512*512 | Rg 660*512 | Ug 660*512
  float* Wsym = (float*)d_ws;
  float* Rg   = Wsym + (size_t)S * S;
  float* Ug   = Rg + (size_t)F * S;

  const int nzero = S * S + F * S;   // Wsym + Rg need zero (atomic accum)
  zero_ws_kernel<<<256, 256, 0, stream>>>((float*)d_ws, nzero);

  dim3 g1(64, NFG);                  // 16x4 tiles  x  20 feature groups
  pass1_kernel<<<g1, 256, 0, stream>>>(tokens, att, weight, Wsym, Rg);

  mid_kernel<<<F, 256, 0, stream>>>(Rg, weight, Ug);

  pass2_kernel<<<128, 256, 0, stream>>>(Ug, Rg, Wsym, bias, out);
}